// GraphAttentionNetwork_78005196030505
// MI455X (gfx1250) — compile-verified
//
#include <hip/hip_runtime.h>
#include <math.h>

#define N_NODES 50000
#define N_EDGESI 800000
#define N_GRAPHS 512
#define DIM 128

typedef __attribute__((ext_vector_type(2))) float v2f;
typedef __attribute__((ext_vector_type(8))) float v8f;

// ---------------- utility ----------------
__global__ void fill_f32(float* __restrict__ p, float v, int n) {
  int i = blockIdx.x * blockDim.x + threadIdx.x;
  int stride = gridDim.x * blockDim.x;
  for (; i < n; i += stride) p[i] = v;
}

__device__ __forceinline__ void atomicMaxF(float* addr, float val) {
  // monotonic float<->int bit trick; addr initialized to -inf
  if (val >= 0.0f) atomicMax((int*)addr, __float_as_int(val));
  else             atomicMin((unsigned int*)addr, __float_as_uint(val));
}

// ---------------- H = X @ W  (M x 128) = (M x 128)(128 x 128), f32 WMMA ----------------
__global__ __launch_bounds__(256)
void gemm_wmma_f32(const float* __restrict__ X, const float* __restrict__ W,
                   float* __restrict__ H, int M) {
  __shared__ float Ws[DIM * DIM];           // 64 KB of the 320 KB/WGP LDS
  for (int i = threadIdx.x; i < DIM * DIM; i += 256) Ws[i] = W[i];
  __syncthreads();

  const int lane = threadIdx.x & 31;
  const int wave = threadIdx.x >> 5;        // 8 waves -> 8 column tiles of 16
  const int n0   = wave * 16;
  const int hi   = lane >> 4;               // 0: lanes 0-15, 1: lanes 16-31
  const int lo   = lane & 15;

  // Preload B fragments for all 32 K-steps (4x16 f32 tile each).
  // Lane striping: lanes 0-15 hold rows K={kb,kb+1}, lanes 16-31 rows K={kb+2,kb+3}.
  v2f bfrag[32];
#pragma unroll
  for (int kk = 0; kk < 32; ++kk) {
    int kb = kk * 4 + hi * 2;
    v2f b;
    b.x = Ws[kb * DIM + n0 + lo];
    b.y = Ws[(kb + 1) * DIM + n0 + lo];
    bfrag[kk] = b;
  }

  const int rbase = blockIdx.x * 64;        // block covers 64 rows
  for (int mt = 0; mt < 4; ++mt) {
    int r0   = rbase + mt * 16;
    int row  = r0 + lo;
    int rowc = row < M ? row : (M - 1);     // clamp; EXEC must stay all-ones for WMMA
    const float* xr = X + rowc * DIM + hi * 2;

    v8f acc = {};
#pragma unroll
    for (int kk = 0; kk < 32; ++kk) {
      v2f a = *(const v2f*)(xr + kk * 4);   // row-contiguous, 8B aligned
      acc = __builtin_amdgcn_wmma_f32_16x16x4_f32(
          /*neg_a=*/false, a, /*neg_b=*/false, bfrag[kk],
          /*c_mod=*/(short)0, acc, /*reuse_a=*/false, /*reuse_b=*/false);
    }
    // C/D layout: VGPR v -> row r0 + v + 8*hi, col n0 + lo
    int col = n0 + lo;
#pragma unroll
    for (int v = 0; v < 8; ++v) {
      int rr = r0 + v + hi * 8;
      if (rr < M) H[rr * DIM + col] = acc[v];
    }
  }
}

// ---------------- per-(node,head) attention logits ----------------
__global__ void alpha_kernel(const float* __restrict__ H, const float* __restrict__ a_src,
                             const float* __restrict__ a_dst, float* __restrict__ as_,
                             float* __restrict__ ad_, int N, int heads, int chan) {
  int idx = blockIdx.x * blockDim.x + threadIdx.x;
  if (idx >= N * heads) return;
  int n = idx / heads, h = idx % heads;
  const float* hp  = H + n * DIM + h * chan;
  const float* av  = a_src + h * chan;
  const float* bv  = a_dst + h * chan;
  float s = 0.f, d = 0.f;
  for (int c = 0; c < chan; ++c) { s = fmaf(hp[c], av[c], s); d = fmaf(hp[c], bv[c], d); }
  as_[idx] = s;
  ad_[idx] = d;
}

// ---------------- segment-softmax pass 1: per-dst max ----------------
__global__ void edge_max_kernel(const int* __restrict__ esrc, const int* __restrict__ edst,
                                const float* __restrict__ as_, const float* __restrict__ ad_,
                                float* __restrict__ m, int heads) {
  int e = blockIdx.x * blockDim.x + threadIdx.x;
  if (e >= N_EDGESI + N_NODES) return;
  int s, d;
  if (e < N_EDGESI) { s = esrc[e]; d = edst[e]; } else { s = d = e - N_EDGESI; }
  for (int h = 0; h < heads; ++h) {
    float v = as_[s * heads + h] + ad_[d * heads + h];
    v = v > 0.f ? v : 0.2f * v;             // leaky_relu(0.2)
    atomicMaxF(&m[d * heads + h], v);
  }
}

// ---------------- pass 2: per-dst denominator ----------------
__global__ void edge_den_kernel(const int* __restrict__ esrc, const int* __restrict__ edst,
                                const float* __restrict__ as_, const float* __restrict__ ad_,
                                const float* __restrict__ m, float* __restrict__ den, int heads) {
  int e = blockIdx.x * blockDim.x + threadIdx.x;
  if (e >= N_EDGESI + N_NODES) return;
  int s, d;
  if (e < N_EDGESI) { s = esrc[e]; d = edst[e]; } else { s = d = e - N_EDGESI; }
  for (int h = 0; h < heads; ++h) {
    float v = as_[s * heads + h] + ad_[d * heads + h];
    v = v > 0.f ? v : 0.2f * v;
    atomicAdd(&den[d * heads + h], __expf(v - m[d * heads + h]));
  }
}

// ---------------- pass 3: weighted aggregation (thread per edge-channel) ----------------
__global__ __launch_bounds__(256)
void edge_agg_kernel(const int* __restrict__ esrc, const int* __restrict__ edst,
                     const float* __restrict__ H, const float* __restrict__ as_,
                     const float* __restrict__ ad_, const float* __restrict__ m,
                     const float* __restrict__ den, float* __restrict__ out,
                     int heads, int chan) {
  int idx = blockIdx.x * blockDim.x + threadIdx.x;
  const int total = (N_EDGESI + N_NODES) * DIM;
  if (idx >= total) return;
  int e = idx >> 7;
  int c = idx & (DIM - 1);
  int s, d;
  if (e < N_EDGESI) { s = esrc[e]; d = edst[e]; } else { s = d = e - N_EDGESI; }
  int h = c / chan;                          // chan=32 (4 heads) or 128 (1 head)
  float v = as_[s * heads + h] + ad_[d * heads + h];
  v = v > 0.f ? v : 0.2f * v;
  float alpha = __expf(v - m[d * heads + h]) / (den[d * heads + h] + 1e-16f);
  atomicAdd(&out[d * DIM + c], alpha * H[s * DIM + c]);
}

// ---------------- bias + (optional) ELU ----------------
__global__ void bias_act_kernel(float* __restrict__ X, const float* __restrict__ b,
                                int n_elems, int do_elu) {
  int idx = blockIdx.x * blockDim.x + threadIdx.x;
  if (idx >= n_elems) return;
  int c = idx & (DIM - 1);
  float v = X[idx] + b[c];
  if (do_elu) v = v > 0.f ? v : (__expf(v) - 1.f);
  X[idx] = v;
}

// ---------------- global mean pool (sum+count) ----------------
__global__ void pool_kernel(const float* __restrict__ X, const int* __restrict__ batch,
                            float* __restrict__ gsum, float* __restrict__ cnt, int N) {
  int idx = blockIdx.x * blockDim.x + threadIdx.x;
  if (idx >= N * DIM) return;
  int n = idx >> 7;
  int c = idx & (DIM - 1);
  int g = batch[n];
  atomicAdd(&gsum[g * DIM + c], X[idx]);
  if (c == 0) atomicAdd(&cnt[g], 1.0f);
}

// ---------------- output MLP: 128 -> 64 (relu) -> 1 ----------------
__global__ __launch_bounds__(64)
void mlp_kernel(const float* __restrict__ gsum, const float* __restrict__ cnt,
                const float* __restrict__ Wo1, const float* __restrict__ bo1,
                const float* __restrict__ Wo2, const float* __restrict__ bo2,
                float* __restrict__ out) {
  __shared__ float g[DIM];
  __shared__ float hid[64];
  int b = blockIdx.x;
  float inv = 1.0f / fmaxf(cnt[b], 1.0f);
  for (int i = threadIdx.x; i < DIM; i += 64) g[i] = gsum[b * DIM + i] * inv;
  __syncthreads();
  int j = threadIdx.x;
  float s = bo1[j];
  for (int c = 0; c < DIM; ++c) s = fmaf(g[c], Wo1[c * 64 + j], s);
  hid[j] = fmaxf(s, 0.0f);
  __syncthreads();
  if (j == 0) {
    float o = bo2[0];
    for (int k = 0; k < 64; ++k) o = fmaf(hid[k], Wo2[k], o);
    out[b] = o;
  }
}

// ---------------- host launcher ----------------
extern "C" void kernel_launch(void* const* d_in, const int* in_sizes, int n_in,
                              void* d_out, int out_size, void* d_ws, size_t ws_size,
                              hipStream_t stream) {
  const float* x      = (const float*)d_in[0];
  const int*   eidx   = (const int*)d_in[1];
  const int*   batch  = (const int*)d_in[2];
  const float* W0     = (const float*)d_in[3];
  const float* as0    = (const float*)d_in[4];
  const float* ad0    = (const float*)d_in[5];
  const float* b0     = (const float*)d_in[6];
  const float* W1     = (const float*)d_in[7];
  const float* as1    = (const float*)d_in[8];
  const float* ad1    = (const float*)d_in[9];
  const float* b1     = (const float*)d_in[10];
  const float* W2     = (const float*)d_in[11];
  const float* as2    = (const float*)d_in[12];
  const float* ad2    = (const float*)d_in[13];
  const float* b2     = (const float*)d_in[14];
  const float* Wo1    = (const float*)d_in[15];
  const float* bo1    = (const float*)d_in[16];
  const float* Wo2    = (const float*)d_in[17];
  const float* bo2    = (const float*)d_in[18];

  const int* esrc = eidx;              // edge_index[0]
  const int* edst = eidx + N_EDGESI;   // edge_index[1]

  float* ws  = (float*)d_ws;
  float* H   = ws;                     // 50000*128
  float* AGG = H   + N_NODES * DIM;    // 50000*128
  float* AS  = AGG + N_NODES * DIM;    // 50000*4
  float* AD  = AS  + N_NODES * 4;
  float* MX  = AD  + N_NODES * 4;
  float* DEN = MX  + N_NODES * 4;
  float* GS  = DEN + N_NODES * 4;      // 512*128
  float* CNT = GS  + N_GRAPHS * DIM;   // 512

  const int E       = N_EDGESI + N_NODES;
  const int ND      = N_NODES * DIM;
  const int gemmGrd = (N_NODES + 63) / 64;

  auto layer = [&](const float* in, const float* Wm, const float* av, const float* dv,
                   const float* bias, int heads, int chan, int do_elu) {
    gemm_wmma_f32<<<gemmGrd, 256, 0, stream>>>(in, Wm, H, N_NODES);
    int nh = N_NODES * heads;
    alpha_kernel<<<(nh + 255) / 256, 256, 0, stream>>>(H, av, dv, AS, AD, N_NODES, heads, chan);
    fill_f32<<<512, 256, 0, stream>>>(MX, -INFINITY, nh);
    fill_f32<<<512, 256, 0, stream>>>(DEN, 0.0f, nh);
    fill_f32<<<2048, 256, 0, stream>>>(AGG, 0.0f, ND);
    edge_max_kernel<<<(E + 255) / 256, 256, 0, stream>>>(esrc, edst, AS, AD, MX, heads);
    edge_den_kernel<<<(E + 255) / 256, 256, 0, stream>>>(esrc, edst, AS, AD, MX, DEN, heads);
    int totEC = E * DIM;
    edge_agg_kernel<<<(totEC + 255) / 256, 256, 0, stream>>>(esrc, edst, H, AS, AD, MX, DEN,
                                                             AGG, heads, chan);
    bias_act_kernel<<<(ND + 255) / 256, 256, 0, stream>>>(AGG, bias, ND, do_elu);
  };

  layer(x,   W0, as0, ad0, b0, 4, 32, 1);
  layer(AGG, W1, as1, ad1, b1, 4, 32, 1);
  layer(AGG, W2, as2, ad2, b2, 1, 128, 0);

  fill_f32<<<64, 256, 0, stream>>>(GS, 0.0f, N_GRAPHS * DIM);
  fill_f32<<<2, 256, 0, stream>>>(CNT, 0.0f, N_GRAPHS);
  pool_kernel<<<(ND + 255) / 256, 256, 0, stream>>>(AGG, batch, GS, CNT, N_NODES);
  mlp_kernel<<<N_GRAPHS, 64, 0, stream>>>(GS, CNT, Wo1, bo1, Wo2, bo2, (float*)d_out);
}